// SOnEquivalentLayer_7103875908198
// MI455X (gfx1250) — compile-verified
//
#include <hip/hip_runtime.h>
#include <hip/hip_bf16.h>
#include <math.h>

// ---------------------------------------------------------------------------
// SO(n)-equivariant message passing layer for MI455X (gfx1250, wave32).
//
//   Phase A: per-edge geometry + Bessel RBF (sin recurrence, 1 sincos/edge)
//   Phase B: per-(edge,channel) messages, rbf_w staged in LDS, scatter via
//            global_atomic_add_f32 into d_out (pre-initialized with residual)
//   Phase C: self-interaction as f32 WMMA (v_wmma_f32_16x16x4_f32):
//            wave = 16 atoms x 16 out-channels, 13 accumulators (one per
//            spatial component) so the norm-gated activation is register-local
// ---------------------------------------------------------------------------

#define C_DIM   64
#define K_RBF   20
#define N_COMBO 11
#define RC_F    5.0f
#define NORM_INV 0.1f
#define EPS_F   1e-12f
#define PI_F    3.14159265358979323846f

typedef float v2f __attribute__((ext_vector_type(2)));
typedef float v8f __attribute__((ext_vector_type(8)));

__device__ __forceinline__ float sigmoidf_(float x) {
  return 1.0f / (1.0f + __expf(-x));
}

__device__ __forceinline__ v8f wmma_f32(v2f a, v2f b, v8f c) {
  // D = A(16x4 f32) * B(4x16 f32) + C(16x16 f32)
  return __builtin_amdgcn_wmma_f32_16x16x4_f32(
      /*neg_a=*/false, a, /*neg_b=*/false, b,
      /*c_mod=*/(short)0, c, /*reuse_a=*/false, /*reuse_b=*/false);
}

// ---------------------------------------------------------------------------
// Phase A: one thread per edge -> unit vector + 20 Bessel RBF features.
// sin(k*t) via recurrence: s_{k+1} = 2cos(t)*s_k - s_{k-1}
// ---------------------------------------------------------------------------
__global__ __launch_bounds__(256) void edge_geom_kernel(
    const float* __restrict__ coords,
    const int*   __restrict__ idx_i,
    const int*   __restrict__ idx_j,
    float*       __restrict__ uE,     // [E,3]
    float*       __restrict__ rbfE,   // [E,20]
    int nEdges)
{
  int e = blockIdx.x * blockDim.x + threadIdx.x;
  if (e >= nEdges) return;
  int i = idx_i[e], j = idx_j[e];
  float rx = coords[j * 3 + 0] - coords[i * 3 + 0];
  float ry = coords[j * 3 + 1] - coords[i * 3 + 1];
  float rz = coords[j * 3 + 2] - coords[i * 3 + 2];
  float d  = sqrtf(rx * rx + ry * ry + rz * rz + EPS_F);
  float inv = 1.0f / d;
  uE[e * 3 + 0] = rx * inv;
  uE[e * 3 + 1] = ry * inv;
  uE[e * 3 + 2] = rz * inv;

  float t = PI_F * d * (1.0f / RC_F);
  float s, c;
  __sincosf(t, &s, &c);
  // smooth cosine cutoff: x = clip(d/RC, 0, 1); fc = 0.5*(cos(pi*x)+1)
  float fc   = (d <= RC_F) ? 0.5f * (c + 1.0f) : 0.0f;
  float pref = sqrtf(2.0f / RC_F) * inv * fc;

  float sp = 0.0f, sc = s, twoc = 2.0f * c;
#pragma unroll
  for (int k = 0; k < K_RBF; ++k) {
    rbfE[e * K_RBF + k] = pref * sc;
    float nx = twoc * sc - sp;
    sp = sc; sc = nx;
  }
}

// ---------------------------------------------------------------------------
// Phase B: one thread per (edge, channel). 4 edges per 256-thread block.
// rbf_w [11][20][64] staged in LDS (56.3 KB). 13 fp32 atomics per thread.
// ---------------------------------------------------------------------------
__global__ __launch_bounds__(256) void edge_message_kernel(
    const float* __restrict__ t0, const float* __restrict__ t1,
    const float* __restrict__ t2,
    const int*   __restrict__ idx_i, const int* __restrict__ idx_j,
    const float* __restrict__ uE, const float* __restrict__ rbfE,
    const float* __restrict__ rbfW,
    float* __restrict__ out0, float* __restrict__ out1,
    float* __restrict__ out2, int nEdges)
{
  __shared__ float wsm[N_COMBO * K_RBF * C_DIM];
  for (int i = threadIdx.x; i < N_COMBO * K_RBF * C_DIM; i += 256)
    wsm[i] = rbfW[i];
  __syncthreads();

  const int gid = blockIdx.x * 256 + threadIdx.x;
  const int e = gid >> 6;
  const int c = gid & 63;
  if (e >= nEdges) return;

  const int ai = idx_i[e], aj = idx_j[e];
  const float u0 = uE[e * 3 + 0], u1 = uE[e * 3 + 1], u2 = uE[e * 3 + 2];
  const float u[3] = {u0, u1, u2};

  float rb[K_RBF];
#pragma unroll
  for (int k = 0; k < K_RBF; ++k) rb[k] = rbfE[e * K_RBF + k];

  // fn[k] = (rbf @ rbf_w[k])[c]; fold in 1/NORM (messages are linear in fn)
  float fn[N_COMBO];
#pragma unroll
  for (int kc = 0; kc < N_COMBO; ++kc) {
    float s = 0.0f;
#pragma unroll
    for (int kk = 0; kk < K_RBF; ++kk)
      s = __fmaf_rn(rb[kk], wsm[(kc * K_RBF + kk) * C_DIM + c], s);
    fn[kc] = s * NORM_INV;
  }

  // gather neighbor features (channel c)
  const float a0 = t0[(size_t)aj * C_DIM + c];
  float a1[3], a2[9];
#pragma unroll
  for (int x = 0; x < 3; ++x) a1[x] = t1[(size_t)aj * 192 + c * 3 + x];
#pragma unroll
  for (int q = 0; q < 9; ++q) a2[q] = t2[(size_t)aj * 576 + c * 9 + q];

  // shared contractions
  const float s1 = a1[0] * u0 + a1[1] * u1 + a1[2] * u2;      // a1 . u
  float w2v[3];
#pragma unroll
  for (int p = 0; p < 3; ++p)
    w2v[p] = a2[p * 3 + 0] * u0 + a2[p * 3 + 1] * u1 + a2[p * 3 + 2] * u2;
  const float q2 = w2v[0] * u0 + w2v[1] * u1 + w2v[2] * u2;   // u^T a2 u

  // way-0: combos (0,0,0), (1,1,0), (2,2,0)
  float m0 = fn[0] * a0 + fn[4] * s1 + fn[9] * q2;
  unsafeAtomicAdd(&out0[(size_t)ai * C_DIM + c], m0);

  // way-1: combos (0,1,1), (1,0,1), (1,2,1), (2,1,1)
#pragma unroll
  for (int p = 0; p < 3; ++p) {
    float m1 = fn[1] * a0 * u[p] + fn[3] * a1[p]
             + fn[6] * s1 * u[p] + fn[8] * w2v[p];
    unsafeAtomicAdd(&out1[(size_t)ai * 192 + c * 3 + p], m1);
  }

  // way-2: combos (0,2,2), (1,1,2), (2,0,2), (2,2,2)
#pragma unroll
  for (int p = 0; p < 3; ++p) {
#pragma unroll
    for (int q = 0; q < 3; ++q) {
      float m2 = fn[2] * a0 * u[p] * u[q] + fn[5] * a1[p] * u[q]
               + fn[7] * a2[p * 3 + q]    + fn[10] * w2v[p] * u[q];
      unsafeAtomicAdd(&out2[(size_t)ai * 576 + c * 9 + p * 3 + q], m2);
    }
  }
}

// ---------------------------------------------------------------------------
// Phase C: self-interaction + 'jilu' activation + residual, in place on d_out.
// Block = 128 threads = 4 waves; block owns 16 atoms, wave owns 16 out-chans.
// LDS layout smem[comp][chan][atom] (13*64*16 f32 = 53 KB).
// WMMA: A = agg rows (16 atoms x K4), B = weight (K4 x 16 outchan), f32.
// ---------------------------------------------------------------------------
__global__ __launch_bounds__(128) void self_interact_kernel(
    float* __restrict__ out0, float* __restrict__ out1,
    float* __restrict__ out2,
    const float* __restrict__ w0, const float* __restrict__ b0,
    const float* __restrict__ w1, const float* __restrict__ w2,
    const float* __restrict__ actw)
{
  __shared__ float smem[13 * C_DIM * 16];
  const int atom0 = blockIdx.x * 16;
  const int tid = threadIdx.x;

  // Stage aggregated features (agg = residual + messages) into LDS.
  for (int g = tid; g < 16 * C_DIM; g += 128) {           // way0
    int m = g >> 6, c = g & 63;
    smem[(0 * C_DIM + c) * 16 + m] = out0[(size_t)(atom0 + m) * C_DIM + c];
  }
  for (int g = tid; g < 16 * C_DIM * 3; g += 128) {       // way1
    int m = g / 192; int r = g % 192; int c = r / 3; int a = r % 3;
    smem[((1 + a) * C_DIM + c) * 16 + m] =
        out1[(size_t)(atom0 + m) * 192 + c * 3 + a];
  }
  for (int g = tid; g < 16 * C_DIM * 9; g += 128) {       // way2
    int m = g / 576; int r = g % 576; int c = r / 9; int pq = r % 9;
    smem[((4 + pq) * C_DIM + c) * 16 + m] =
        out2[(size_t)(atom0 + m) * 576 + c * 9 + pq];
  }
  __syncthreads();

  const int lane = tid & 31;
  const int wave = tid >> 5;
  const int col0 = wave * 16;
  const int half = lane >> 4;       // 0: K pair {k0,k0+1}, 1: {k0+2,k0+3}
  const int mlow = lane & 15;
  const int col  = col0 + mlow;     // out channel for B/C/D fragments

  v8f acc[13] = {};                 // [comp]: 0=scalar, 1-3=vector, 4-12=tensor

#pragma unroll
  for (int ks = 0; ks < 16; ++ks) {
    const int kb = ks * 4 + half * 2;
    // B fragments: W[k][o], rows striped across lanes (K pair per half-wave)
    v2f bw0, bw1, bw2;
    bw0.x = w0[kb * C_DIM + col];  bw0.y = w0[(kb + 1) * C_DIM + col];
    bw1.x = w1[kb * C_DIM + col];  bw1.y = w1[(kb + 1) * C_DIM + col];
    bw2.x = w2[kb * C_DIM + col];  bw2.y = w2[(kb + 1) * C_DIM + col];
#pragma unroll
    for (int comp = 0; comp < 13; ++comp) {
      v2f a;  // A fragment: lanes 0-15 = rows M, K pair per half-wave
      a.x = smem[(comp * C_DIM + kb) * 16 + mlow];
      a.y = smem[(comp * C_DIM + kb + 1) * 16 + mlow];
      v2f b = (comp == 0) ? bw0 : ((comp <= 3) ? bw1 : bw2);
      acc[comp] = wmma_f32(a, b, acc[comp]);
    }
  }

  const float bias = b0[col];
  const float aw0 = actw[col], aw1 = actw[64 + col], aw2 = actw[128 + col];

  // C/D layout: VGPR r -> rows {r, r+8} for lane halves, N = lane&15
#pragma unroll
  for (int r = 0; r < 8; ++r) {
    const int m = half ? (r + 8) : r;
    const size_t atom = (size_t)(atom0 + m);

    // way 0: learnable swish
    {
      float y = acc[0][r] + bias;
      float z = y * sigmoidf_(aw0 * y);
      out0[atom * C_DIM + col] = z + smem[(0 * C_DIM + col) * 16 + m];
    }
    // way 1: norm-gated sigmoid over 3 components
    {
      float v0 = acc[1][r], v1 = acc[2][r], v2 = acc[3][r];
      float n = sqrtf(v0 * v0 + v1 * v1 + v2 * v2 + EPS_F);
      float g = sigmoidf_(aw1 * n);
      out1[atom * 192 + col * 3 + 0] = v0 * g + smem[(1 * C_DIM + col) * 16 + m];
      out1[atom * 192 + col * 3 + 1] = v1 * g + smem[(2 * C_DIM + col) * 16 + m];
      out1[atom * 192 + col * 3 + 2] = v2 * g + smem[(3 * C_DIM + col) * 16 + m];
    }
    // way 2: norm-gated sigmoid over 9 components
    {
      float vv[9]; float ss = EPS_F;
#pragma unroll
      for (int q = 0; q < 9; ++q) { vv[q] = acc[4 + q][r]; ss += vv[q] * vv[q]; }
      float n = sqrtf(ss);
      float g = sigmoidf_(aw2 * n);
#pragma unroll
      for (int q = 0; q < 9; ++q)
        out2[atom * 576 + col * 9 + q] =
            vv[q] * g + smem[((4 + q) * C_DIM + col) * 16 + m];
    }
  }
}

// ---------------------------------------------------------------------------
extern "C" void kernel_launch(void* const* d_in, const int* in_sizes, int n_in,
                              void* d_out, int out_size, void* d_ws, size_t ws_size,
                              hipStream_t stream) {
  const float* t0     = (const float*)d_in[0];
  const float* t1     = (const float*)d_in[1];
  const float* t2     = (const float*)d_in[2];
  const float* coords = (const float*)d_in[3];
  const int*   eidx   = (const int*)  d_in[4];
  const float* rbfW   = (const float*)d_in[5];
  const float* w0     = (const float*)d_in[6];
  const float* b0     = (const float*)d_in[7];
  const float* w1     = (const float*)d_in[8];
  const float* w2     = (const float*)d_in[9];
  const float* actw   = (const float*)d_in[10];

  const int nAtoms = in_sizes[0] / C_DIM;
  const int nEdges = in_sizes[4] / 2;
  const int* idx_i = eidx;
  const int* idx_j = eidx + nEdges;

  float* out0 = (float*)d_out;
  float* out1 = out0 + (size_t)nAtoms * C_DIM;
  float* out2 = out1 + (size_t)nAtoms * C_DIM * 3;

  float* uE   = (float*)d_ws;                   // [E,3]
  float* rbfE = uE + (size_t)3 * nEdges;        // [E,20]  (total ~18.4 MB)

  // residual init: agg <- inputs (d_out is the aggregation buffer)
  hipMemcpyAsync(out0, t0, (size_t)nAtoms * C_DIM * sizeof(float),
                 hipMemcpyDeviceToDevice, stream);
  hipMemcpyAsync(out1, t1, (size_t)nAtoms * C_DIM * 3 * sizeof(float),
                 hipMemcpyDeviceToDevice, stream);
  hipMemcpyAsync(out2, t2, (size_t)nAtoms * C_DIM * 9 * sizeof(float),
                 hipMemcpyDeviceToDevice, stream);

  edge_geom_kernel<<<(nEdges + 255) / 256, 256, 0, stream>>>(
      coords, idx_i, idx_j, uE, rbfE, nEdges);

  long totB = (long)nEdges * C_DIM;
  edge_message_kernel<<<(int)((totB + 255) / 256), 256, 0, stream>>>(
      t0, t1, t2, idx_i, idx_j, uE, rbfE, rbfW, out0, out1, out2, nEdges);

  self_interact_kernel<<<nAtoms / 16, 128, 0, stream>>>(
      out0, out1, out2, w0, b0, w1, w2, actw);
}